// sPHNNDynamicsFixed_45612552684107
// MI455X (gfx1250) — compile-verified
//
#include <hip/hip_runtime.h>
#include <math.h>

// ---------------------------------------------------------------------------
// Types for CDNA5 f32 WMMA (V_WMMA_F32_16X16X4_F32):
//   A: 16x4  f32 -> 2 VGPRs/lane   (lane = M row; lanes 0-15 hold K=0,1; lanes 16-31 hold K=2,3)
//   B: 4x16  f32 -> 2 VGPRs/lane   (lane = N col; lanes 0-15 hold K=0,1; lanes 16-31 hold K=2,3)
//   C/D: 16x16 f32 -> 8 VGPRs/lane (reg v: M=v (lanes 0-15) / M=v+8 (lanes 16-31); N=lane%16)
// ---------------------------------------------------------------------------
typedef __attribute__((ext_vector_type(2))) float v2f;
typedef __attribute__((ext_vector_type(8))) float v8f;

static __device__ __forceinline__ v8f wmma4(v2f a, v2f b, v8f c) {
  // 8 args: (neg_a, A, neg_b, B, c_mod, C, reuse_a, reuse_b)
  return __builtin_amdgcn_wmma_f32_16x16x4_f32(false, a, false, b, (short)0, c,
                                               false, false);
}

static __device__ __forceinline__ float spf(float x) {
  // numerically-stable softplus, matches jax.nn.softplus = logaddexp(x, 0)
  return fmaxf(x, 0.f) + log1pf(expf(-fabsf(x)));
}

// ---------------------------------------------------------------------------
// Kernel 1: per-timestep argmax over src column, emit f = [pos0,pos1,pos2,src]
// ---------------------------------------------------------------------------
__global__ void __launch_bounds__(256) k_argmax(const float* __restrict__ nodes,
                                                float* __restrict__ Fbuf,
                                                int Nnodes) {
  const int t = blockIdx.x;
  const float* base = nodes + (size_t)t * (size_t)Nnodes * 5u;
  const int tid = threadIdx.x;

  float best = -INFINITY;
  int bidx = 0x7fffffff;
  for (int i = tid; i < Nnodes; i += 256) {
    float v = base[(size_t)i * 5u + 4u];
    if (v > best) { best = v; bidx = i; }  // in-thread visits i ascending -> first occurrence
  }

  __shared__ float sv[256];
  __shared__ int si[256];
  sv[tid] = best;
  si[tid] = bidx;
  __syncthreads();
  for (int s = 128; s > 0; s >>= 1) {
    if (tid < s) {
      float ov = sv[tid + s];
      int oi = si[tid + s];
      if (ov > sv[tid] || (ov == sv[tid] && oi < si[tid])) {
        sv[tid] = ov; si[tid] = oi;
      }
    }
    __syncthreads();
  }
  if (tid == 0) {
    size_t i = (size_t)si[0] * 5u;
    Fbuf[t * 4 + 0] = base[i + 0];
    Fbuf[t * 4 + 1] = base[i + 1];
    Fbuf[t * 4 + 2] = base[i + 2];
    Fbuf[t * 4 + 3] = base[i + 4];
  }
}

// ---------------------------------------------------------------------------
// Kernel 2: WMMA encoder chain per 16-timestep tile:
//   H0 = lrelu(F @ W0^T + b0)        1  WMMA (K=4)
//   H1 = lrelu(H0 @ W1^T + b1)       4  WMMAs (K=16)
//   U  = H1 @ W2pad^T + b2           4  WMMAs (K=16, N-cols 8..15 zero)
//   Bu = U @ Bmat^T  (16x8 @ 8x32)   4  WMMAs (2 N-halves x K=8)
// ---------------------------------------------------------------------------
__global__ void __launch_bounds__(32) k_encoder(
    const float* __restrict__ Fbuf, const float* __restrict__ W0,
    const float* __restrict__ b0v, const float* __restrict__ W1,
    const float* __restrict__ b1v, const float* __restrict__ W2,
    const float* __restrict__ b2v, const float* __restrict__ Bm,
    float* __restrict__ Bu) {
  const int lane = threadIdx.x;
  const bool hi = lane >= 16;
  const int l = lane & 15;
  const int t0 = blockIdx.x * 16;
  const int k0 = hi ? 2 : 0;         // first K index held by this lane in each 4-chunk
  const int mofs = hi ? 8 : 0;       // D-layout row offset

  __shared__ float sh[16 * 16];

  // ---- layer 0: A = F tile (16x4), B = W0^T (4x16) ----
  v2f a0;
  a0.x = Fbuf[(t0 + l) * 4 + k0];
  a0.y = Fbuf[(t0 + l) * 4 + k0 + 1];
  v2f bw0;
  bw0.x = W0[l * 4 + k0];            // B[k][n] = W0[n][k], n = l
  bw0.y = W0[l * 4 + k0 + 1];
  v8f c0 = {0.f, 0.f, 0.f, 0.f, 0.f, 0.f, 0.f, 0.f};
  c0 = wmma4(a0, bw0, c0);
  {
    float bb = b0v[l];
#pragma unroll
    for (int v = 0; v < 8; ++v) {
      float x = c0[v] + bb;
      c0[v] = x >= 0.f ? x : 0.01f * x;
    }
  }
#pragma unroll
  for (int v = 0; v < 8; ++v) sh[(v + mofs) * 16 + l] = c0[v];
  __syncthreads();

  // ---- layer 1: K=16, A from LDS transpose of H0, B = W1^T ----
  v2f a1[4], bw1[4];
#pragma unroll
  for (int kk = 0; kk < 4; ++kk) {
    int k = 4 * kk + k0;
    a1[kk].x = sh[l * 16 + k];
    a1[kk].y = sh[l * 16 + k + 1];
    bw1[kk].x = W1[l * 16 + k];      // B[k][n] = W1[n][k]
    bw1[kk].y = W1[l * 16 + k + 1];
  }
  __syncthreads();
  v8f c1 = {0.f, 0.f, 0.f, 0.f, 0.f, 0.f, 0.f, 0.f};
#pragma unroll
  for (int kk = 0; kk < 4; ++kk) c1 = wmma4(a1[kk], bw1[kk], c1);
  {
    float bb = b1v[l];
#pragma unroll
    for (int v = 0; v < 8; ++v) {
      float x = c1[v] + bb;
      c1[v] = x >= 0.f ? x : 0.01f * x;
    }
  }
#pragma unroll
  for (int v = 0; v < 8; ++v) sh[(v + mofs) * 16 + l] = c1[v];
  __syncthreads();

  // ---- layer 2: K=16, B = W2^T zero-padded to 16 columns ----
  v2f a2[4], bw2[4];
#pragma unroll
  for (int kk = 0; kk < 4; ++kk) {
    int k = 4 * kk + k0;
    a2[kk].x = sh[l * 16 + k];
    a2[kk].y = sh[l * 16 + k + 1];
    bw2[kk].x = (l < 8) ? W2[l * 16 + k] : 0.f;       // B[k][n] = W2[n][k], n<8
    bw2[kk].y = (l < 8) ? W2[l * 16 + k + 1] : 0.f;
  }
  __syncthreads();
  v8f c2 = {0.f, 0.f, 0.f, 0.f, 0.f, 0.f, 0.f, 0.f};
#pragma unroll
  for (int kk = 0; kk < 4; ++kk) c2 = wmma4(a2[kk], bw2[kk], c2);
  {
    float bb = (l < 8) ? b2v[l] : 0.f;
#pragma unroll
    for (int v = 0; v < 8; ++v) c2[v] += bb;
  }
#pragma unroll
  for (int v = 0; v < 8; ++v) sh[(v + mofs) * 16 + l] = c2[v];
  __syncthreads();

  // ---- Bu = U(16x8) @ Bmat^T(8x32): two 16-col halves, K=8 (2 chunks) ----
  v2f a3[2];
#pragma unroll
  for (int kk = 0; kk < 2; ++kk) {
    int k = 4 * kk + k0;
    a3[kk].x = sh[l * 16 + k];
    a3[kk].y = sh[l * 16 + k + 1];
  }
#pragma unroll
  for (int half = 0; half < 2; ++half) {
    int s = half * 16 + l;           // state dim handled by this lane's N column
    v2f bb0, bb1;
    bb0.x = Bm[s * 8 + k0];          // B[k][n'] = Bmat[s][k]
    bb0.y = Bm[s * 8 + k0 + 1];
    bb1.x = Bm[s * 8 + 4 + k0];
    bb1.y = Bm[s * 8 + 4 + k0 + 1];
    v8f cb = {0.f, 0.f, 0.f, 0.f, 0.f, 0.f, 0.f, 0.f};
    cb = wmma4(a3[0], bb0, cb);
    cb = wmma4(a3[1], bb1, cb);
#pragma unroll
    for (int v = 0; v < 8; ++v) Bu[(t0 + v + mofs) * 32 + s] = cb[v];
  }
}

// ---------------------------------------------------------------------------
// Kernel 3: sequential RK4 scan. One wave; lane j owns state dim j.
// ---------------------------------------------------------------------------
__global__ void __launch_bounds__(32) k_scan(
    const float* __restrict__ ts, const float* __restrict__ z0,
    const float* __restrict__ Wx0, const float* __restrict__ fb0,
    const float* __restrict__ Wz1, const float* __restrict__ Wx1,
    const float* __restrict__ fb1, const float* __restrict__ Wz2,
    const float* __restrict__ Wx2, const float* __restrict__ J,
    const float* __restrict__ R, const float* __restrict__ Bu,
    float* __restrict__ out, int T) {
  const int lane = threadIdx.x;  // 0..31
  const bool lo = lane < 16;
  const int li = lane & 15;

  __shared__ float tsL[1024];
  __shared__ float Jl[1024];
  __shared__ float Rl[1024];
  __shared__ float Atmp[1024];
  __shared__ float zL[32], hL[16], dLs[16], eLs[16], gL[32];

#pragma unroll 1
  for (int i = lane; i < 1024; i += 32) {
    Jl[i] = J[i];
    Rl[i] = R[i];
  }
#pragma unroll 1
  for (int i = lane; i < T; i += 32) tsL[i] = ts[i];
  __syncthreads();

  // ---- A = J - J^T - R R^T, row `lane` into registers ----
  {
    float rj[32];
#pragma unroll
    for (int m = 0; m < 32; ++m) rj[m] = Rl[lane * 32 + m];
#pragma unroll 1
    for (int k = 0; k < 32; ++k) {
      float s0 = 0.f, s1 = 0.f, s2 = 0.f, s3 = 0.f;
#pragma unroll
      for (int m = 0; m < 32; m += 4) {
        s0 += rj[m] * Rl[k * 32 + m];
        s1 += rj[m + 1] * Rl[k * 32 + m + 1];
        s2 += rj[m + 2] * Rl[k * 32 + m + 2];
        s3 += rj[m + 3] * Rl[k * 32 + m + 3];
      }
      Atmp[lane * 32 + k] =
          Jl[lane * 32 + k] - Jl[k * 32 + lane] - ((s0 + s1) + (s2 + s3));
    }
  }
  __syncthreads();
  float Ar[32];
#pragma unroll
  for (int k = 0; k < 32; ++k) Ar[k] = Atmp[lane * 32 + k];

  // ---- preload weights into per-lane registers ----
  float wx0r[32], wx1r[32], wz1r[16], wz1t[16], wx0t[16], wx1t[16];
#pragma unroll
  for (int i = 0; i < 32; ++i) wx0r[i] = Wx0[li * 32 + i];   // row li (used by lanes<16)
#pragma unroll
  for (int i = 0; i < 32; ++i) wx1r[i] = Wx1[li * 32 + i];
#pragma unroll
  for (int k = 0; k < 16; ++k) wz1r[k] = spf(Wz1[li * 16 + k]);       // sp(Wz1) row
#pragma unroll
  for (int k = 0; k < 16; ++k) wz1t[k] = spf(Wz1[k * 16 + li]);       // sp(Wz1)^T row
#pragma unroll
  for (int i = 0; i < 16; ++i) wx0t[i] = Wx0[i * 32 + lane];          // Wx0^T row lane
#pragma unroll
  for (int i = 0; i < 16; ++i) wx1t[i] = Wx1[i * 32 + lane];
  const float spwz2 = spf(Wz2[li]);
  const float b0l = fb0[li];
  const float b1l = fb1[li];
  const float wx2l = Wx2[lane];

  // ---- deriv(z, Bu): A @ gradH(z) + Bu ----
  auto deriv = [&](float zj, float buj) -> float {
    zL[lane] = zj;
    __syncthreads();
    float pre0, p1x;
    {
      float a0 = 0.f, a1 = 0.f, a2 = 0.f, a3 = 0.f;
      float c0 = 0.f, c1 = 0.f, c2 = 0.f, c3 = 0.f;
#pragma unroll
      for (int j = 0; j < 32; j += 4) {
        float v0 = zL[j], v1 = zL[j + 1], v2 = zL[j + 2], v3 = zL[j + 3];
        a0 += wx0r[j] * v0; a1 += wx0r[j + 1] * v1;
        a2 += wx0r[j + 2] * v2; a3 += wx0r[j + 3] * v3;
        c0 += wx1r[j] * v0; c1 += wx1r[j + 1] * v1;
        c2 += wx1r[j + 2] * v2; c3 += wx1r[j + 3] * v3;
      }
      pre0 = b0l + ((a0 + a1) + (a2 + a3));
      p1x = ((c0 + c1) + (c2 + c3));
    }
    float s0 = 0.f;
    if (lo) {
      float e0 = expf(-fabsf(pre0));
      float inv = 1.f / (1.f + e0);
      s0 = (pre0 >= 0.f) ? inv : e0 * inv;               // sigmoid(pre0)
      hL[lane] = fmaxf(pre0, 0.f) + log1pf(e0);          // softplus(pre0)
    }
    __syncthreads();
    if (lo) {
      float p0 = 0.f, p1 = 0.f;
#pragma unroll
      for (int k = 0; k < 16; k += 2) {
        p0 += wz1r[k] * hL[k];
        p1 += wz1r[k + 1] * hL[k + 1];
      }
      float pre1 = b1l + p1x + (p0 + p1);
      float s1 = 1.f / (1.f + expf(-pre1));
      dLs[lane] = s1 * spwz2;                            // dH/dpre1
    }
    __syncthreads();
    if (lo) {
      float q0 = 0.f, q1 = 0.f;
#pragma unroll
      for (int k = 0; k < 16; k += 2) {
        q0 += wz1t[k] * dLs[k];
        q1 += wz1t[k + 1] * dLs[k + 1];
      }
      eLs[lane] = s0 * (q0 + q1);                        // dH/dpre0
    }
    __syncthreads();
    {
      float g0 = 0.f, g1 = 0.f;
#pragma unroll
      for (int i = 0; i < 16; i += 2) {
        g0 += wx0t[i] * eLs[i] + wx1t[i] * dLs[i];
        g1 += wx0t[i + 1] * eLs[i + 1] + wx1t[i + 1] * dLs[i + 1];
      }
      gL[lane] = wx2l + (g0 + g1);                       // gradH
    }
    __syncthreads();
    float d0 = 0.f, d1 = 0.f, d2 = 0.f, d3 = 0.f;
#pragma unroll
    for (int k = 0; k < 32; k += 4) {
      d0 += Ar[k] * gL[k];
      d1 += Ar[k + 1] * gL[k + 1];
      d2 += Ar[k + 2] * gL[k + 2];
      d3 += Ar[k + 3] * gL[k + 3];
    }
    return buj + ((d0 + d1) + (d2 + d3));
  };

  float z = z0[lane];
  out[lane] = z;
#pragma unroll 1
  for (int t = 0; t < T - 1; ++t) {
    float dt = tsL[t + 1] - tsL[t];
    float buj = Bu[t * 32 + lane];
    float k1 = deriv(z, buj);
    float k2 = deriv(z + 0.5f * dt * k1, buj);
    float k3 = deriv(z + 0.5f * dt * k2, buj);
    float k4 = deriv(z + dt * k3, buj);
    z = z + dt / 6.f * (k1 + 2.f * k2 + 2.f * k3 + k4);
    out[(t + 1) * 32 + lane] = z;
  }
}

// ---------------------------------------------------------------------------
extern "C" void kernel_launch(void* const* d_in, const int* in_sizes, int n_in,
                              void* d_out, int out_size, void* d_ws,
                              size_t ws_size, hipStream_t stream) {
  (void)n_in; (void)out_size; (void)ws_size;
  const float* ts = (const float*)d_in[0];
  const float* z0 = (const float*)d_in[1];
  const float* nodes = (const float*)d_in[2];
  const float* eW0 = (const float*)d_in[3];
  const float* eb0 = (const float*)d_in[4];
  const float* eW1 = (const float*)d_in[5];
  const float* eb1 = (const float*)d_in[6];
  const float* eW2 = (const float*)d_in[7];
  const float* eb2 = (const float*)d_in[8];
  const float* Wx0 = (const float*)d_in[9];
  const float* fb0 = (const float*)d_in[10];
  const float* Wz1 = (const float*)d_in[11];
  const float* Wx1 = (const float*)d_in[12];
  const float* fb1 = (const float*)d_in[13];
  const float* Wz2 = (const float*)d_in[14];
  const float* Wx2 = (const float*)d_in[15];
  // d_in[16] = fic_b2: constant offset, vanishes in grad(H) -> unused
  const float* J = (const float*)d_in[17];
  const float* R = (const float*)d_in[18];
  const float* Bm = (const float*)d_in[19];

  const int T = in_sizes[0];                  // 1024
  const int Nnodes = in_sizes[2] / (T * 5);   // 8192

  float* out = (float*)d_out;
  float* Fbuf = (float*)d_ws;                 // T x 4
  float* Bu = Fbuf + (size_t)T * 4;           // T x 32

  k_argmax<<<T, 256, 0, stream>>>(nodes, Fbuf, Nnodes);
  k_encoder<<<T / 16, 32, 0, stream>>>(Fbuf, eW0, eb0, eW1, eb1, eW2, eb2, Bm,
                                       Bu);
  k_scan<<<1, 32, 0, stream>>>(ts, z0, Wx0, fb0, Wz1, Wx1, fb1, Wz2, Wx2, J, R,
                               Bu, out, T);
}